// EncoderObjsAttrsAverage_51951924413027
// MI455X (gfx1250) — compile-verified
//
#include <hip/hip_runtime.h>
#include <hip/hip_bf16.h>

typedef __attribute__((ext_vector_type(2))) float v2f;
typedef __attribute__((ext_vector_type(8))) float v8f;

#define SEG_B 8192
#define ROWS_N 524288
#define DIM_D 256
#define DIM_E 512
#define BN_EPS 1e-5f

// ---------------------------------------------------------------------------
// Kernel 1: exclusive prefix sum of lengths[8192] -> offsets[8192]
// Single block, 1024 threads, 8 elements/thread, LDS Hillis-Steele scan.
// ---------------------------------------------------------------------------
__global__ void excl_scan_kernel(const int* __restrict__ lengths,
                                 int* __restrict__ offsets) {
    __shared__ int s[1024];
    const int t = threadIdx.x;
    const int base = t * 8;
    int l[8];
    int sum = 0;
#pragma unroll
    for (int i = 0; i < 8; ++i) { l[i] = lengths[base + i]; sum += l[i]; }
    s[t] = sum;
    __syncthreads();
    for (int off = 1; off < 1024; off <<= 1) {
        int v = (t >= off) ? s[t - off] : 0;
        __syncthreads();
        s[t] += v;
        __syncthreads();
    }
    int ex = (t > 0) ? s[t - 1] : 0;
#pragma unroll
    for (int i = 0; i < 8; ++i) { offsets[base + i] = ex; ex += l[i]; }
}

// ---------------------------------------------------------------------------
// Kernel 2: segment mean. One block per segment, 64 threads, each thread owns
// 4 contiguous columns -> one global_load_b128 per thread per row (1KB/row
// coalesced), 4-row unroll = 64B in flight per thread. This pass streams the
// 512MB input and sets the runtime floor (~22us at 23.3 TB/s).
// ---------------------------------------------------------------------------
__global__ void seg_mean_kernel(const float* __restrict__ attrs,
                                const int* __restrict__ lengths,
                                const int* __restrict__ offsets,
                                float* __restrict__ avg) {
    const int b = blockIdx.x;
    const int t = threadIdx.x;              // 0..63, owns cols 4t..4t+3
    const int len = lengths[b];
    const int off = offsets[b];
    const float4* p = (const float4*)(attrs + (size_t)off * DIM_D) + t;
    // row stride in float4 units = DIM_D/4 = 64
    float4 a0 = {0.f, 0.f, 0.f, 0.f};
    float4 a1 = {0.f, 0.f, 0.f, 0.f};
    float4 a2 = {0.f, 0.f, 0.f, 0.f};
    float4 a3 = {0.f, 0.f, 0.f, 0.f};
    int r = 0;
    for (; r + 4 <= len; r += 4) {
        float4 v0 = p[(size_t)(r + 0) * 64];
        float4 v1 = p[(size_t)(r + 1) * 64];
        float4 v2 = p[(size_t)(r + 2) * 64];
        float4 v3 = p[(size_t)(r + 3) * 64];
        a0.x += v0.x; a0.y += v0.y; a0.z += v0.z; a0.w += v0.w;
        a1.x += v1.x; a1.y += v1.y; a1.z += v1.z; a1.w += v1.w;
        a2.x += v2.x; a2.y += v2.y; a2.z += v2.z; a2.w += v2.w;
        a3.x += v3.x; a3.y += v3.y; a3.z += v3.z; a3.w += v3.w;
    }
    for (; r < len; ++r) {
        float4 v = p[(size_t)r * 64];
        a0.x += v.x; a0.y += v.y; a0.z += v.z; a0.w += v.w;
    }
    float4 acc;
    acc.x = (a0.x + a1.x) + (a2.x + a3.x);
    acc.y = (a0.y + a1.y) + (a2.y + a3.y);
    acc.z = (a0.z + a1.z) + (a2.z + a3.z);
    acc.w = (a0.w + a1.w) + (a2.w + a3.w);
    float4 out = {0.f, 0.f, 0.f, 0.f};
    if (len > 0) {
        const float inv = 1.f / (float)len;
        out.x = acc.x * inv; out.y = acc.y * inv;
        out.z = acc.z * inv; out.w = acc.w * inv;
    }
    ((float4*)(avg + (size_t)b * DIM_D))[t] = out;
}

// ---------------------------------------------------------------------------
// Kernel 3: x = averages @ W^T + bias via V_WMMA_F32_16X16X4_F32.
// Register-blocked: one wave computes a 64x32 output region (4 M-tiles x
// 2 N-tiles) -> per k-step: 4 A-frag + 2 B-frag loads feed 8 WMMAs
// (0.75 loads/wmma vs 2.0 unblocked; L2 fragment traffic 512MB -> 192MB).
// A-frag (16x4 f32): lane (half=l>>4, m=l&15): v0=A[m][k+2h], v1=A[m][k+2h+1]
// B-frag (4x16 f32): Bmat[k][n] = W[e0+n][k]: v0=W[n][k+2h], v1=W[n][k+2h+1]
// C/D (16x16 f32): c[r] -> row +r+8*half, col idx.
// ---------------------------------------------------------------------------
__global__ void gemm_wmma_kernel(const float* __restrict__ avg,
                                 const float* __restrict__ W,
                                 const float* __restrict__ bias,
                                 float* __restrict__ x) {
    const int wavesPerBlock = blockDim.x >> 5;
    const int wave = blockIdx.x * wavesPerBlock + (threadIdx.x >> 5);
    const int lane = threadIdx.x & 31;
    const int tilesN = DIM_E / 32;          // 16 (N-blocks of 32)
    const int tileN = wave % tilesN;
    const int tileM = wave / tilesN;        // 0..127 (M-blocks of 64)
    const int m0 = tileM * 64;
    const int e0 = tileN * 32;
    const int half = lane >> 4;             // selects K pair {0,1} vs {2,3}
    const int idx = lane & 15;

    const float* a0p = avg + (size_t)(m0 + 0  + idx) * DIM_D + 2 * half;
    const float* a1p = avg + (size_t)(m0 + 16 + idx) * DIM_D + 2 * half;
    const float* a2p = avg + (size_t)(m0 + 32 + idx) * DIM_D + 2 * half;
    const float* a3p = avg + (size_t)(m0 + 48 + idx) * DIM_D + 2 * half;
    const float* b0p = W + (size_t)(e0 + 0  + idx) * DIM_D + 2 * half;
    const float* b1p = W + (size_t)(e0 + 16 + idx) * DIM_D + 2 * half;

    v8f c00 = {}, c01 = {}, c10 = {}, c11 = {};
    v8f c20 = {}, c21 = {}, c30 = {}, c31 = {};
#pragma unroll 2
    for (int k = 0; k < DIM_D; k += 4) {
        v2f a0 = *(const v2f*)(a0p + k);
        v2f a1 = *(const v2f*)(a1p + k);
        v2f a2 = *(const v2f*)(a2p + k);
        v2f a3 = *(const v2f*)(a3p + k);
        v2f b0 = *(const v2f*)(b0p + k);
        v2f b1 = *(const v2f*)(b1p + k);
        c00 = __builtin_amdgcn_wmma_f32_16x16x4_f32(false, a0, false, b0, (short)0, c00, false, false);
        c01 = __builtin_amdgcn_wmma_f32_16x16x4_f32(false, a0, false, b1, (short)0, c01, false, false);
        c10 = __builtin_amdgcn_wmma_f32_16x16x4_f32(false, a1, false, b0, (short)0, c10, false, false);
        c11 = __builtin_amdgcn_wmma_f32_16x16x4_f32(false, a1, false, b1, (short)0, c11, false, false);
        c20 = __builtin_amdgcn_wmma_f32_16x16x4_f32(false, a2, false, b0, (short)0, c20, false, false);
        c21 = __builtin_amdgcn_wmma_f32_16x16x4_f32(false, a2, false, b1, (short)0, c21, false, false);
        c30 = __builtin_amdgcn_wmma_f32_16x16x4_f32(false, a3, false, b0, (short)0, c30, false, false);
        c31 = __builtin_amdgcn_wmma_f32_16x16x4_f32(false, a3, false, b1, (short)0, c31, false, false);
    }

    const float bv0 = bias[e0 + 0  + idx];
    const float bv1 = bias[e0 + 16 + idx];
    const v8f* cs[4][2] = {{&c00, &c01}, {&c10, &c11}, {&c20, &c21}, {&c30, &c31}};
#pragma unroll
    for (int i = 0; i < 4; ++i) {
#pragma unroll
        for (int r = 0; r < 8; ++r) {
            const int m = m0 + 16 * i + r + 8 * half;
            float* row = x + (size_t)m * DIM_E + e0 + idx;
            row[0]  = (*cs[i][0])[r] + bv0;
            row[16] = (*cs[i][1])[r] + bv1;
        }
    }
}

// ---------------------------------------------------------------------------
// Kernel 4a: zero the per-column accumulators (sums, sumsq) -> 1024 floats.
// ---------------------------------------------------------------------------
__global__ void zero_kernel(float* __restrict__ p, int n) {
    int i = blockIdx.x * blockDim.x + threadIdx.x;
    if (i < n) p[i] = 0.f;
}

// ---------------------------------------------------------------------------
// Kernel 4b: BN column stats. grid (32 row-chunks x 2 col-chunks), 256 thr.
// Each block accumulates 256 rows x 256 cols (coalesced) then atomicAdd.
// ---------------------------------------------------------------------------
__global__ void bn_stats_kernel(const float* __restrict__ x,
                                float* __restrict__ sums,
                                float* __restrict__ sumsq) {
    const int e = blockIdx.y * 256 + threadIdx.x;
    const int r0 = blockIdx.x * 256;
    float s = 0.f, sq = 0.f;
#pragma unroll 4
    for (int r = 0; r < 256; ++r) {
        float v = x[(size_t)(r0 + r) * DIM_E + e];
        s += v;
        sq += v * v;
    }
    atomicAdd(&sums[e], s);
    atomicAdd(&sumsq[e], sq);
}

// ---------------------------------------------------------------------------
// Kernel 5: BN normalize + affine. out = (x - mu) * rsqrt(var+eps)*g + b.
// mu = sum/B, var = sumsq/B - mu^2 (biased, matches reference).
// ---------------------------------------------------------------------------
__global__ void bn_apply_kernel(const float* __restrict__ x,
                                const float* __restrict__ sums,
                                const float* __restrict__ sumsq,
                                const float* __restrict__ gamma,
                                const float* __restrict__ beta,
                                float* __restrict__ out) {
    const int i = blockIdx.x * blockDim.x + threadIdx.x;
    const int e = i & (DIM_E - 1);
    const float inv_b = 1.f / (float)SEG_B;
    const float mu = sums[e] * inv_b;
    const float var = sumsq[e] * inv_b - mu * mu;
    const float rs = rsqrtf(var + BN_EPS);
    out[i] = (x[i] - mu) * rs * gamma[e] + beta[e];
}

// ---------------------------------------------------------------------------
extern "C" void kernel_launch(void* const* d_in, const int* in_sizes, int n_in,
                              void* d_out, int out_size, void* d_ws, size_t ws_size,
                              hipStream_t stream) {
    const float* objs_attrs = (const float*)d_in[0];  // [N, D]
    const float* weight     = (const float*)d_in[1];  // [E, D]
    const float* bias       = (const float*)d_in[2];  // [E]
    const float* gamma      = (const float*)d_in[3];  // [E]
    const float* beta       = (const float*)d_in[4];  // [E]
    const int*   lengths    = (const int*)d_in[5];    // [B]

    char* ws = (char*)d_ws;
    // workspace layout (256B aligned):
    //   offsets : 8192 * 4            =    32768 B @ 0
    //   avg     : 8192 * 256 * 4      =  8388608 B @ 32768
    //   x       : 8192 * 512 * 4      = 16777216 B @ 8421376
    //   sums    : 512 * 4             =     2048 B @ 25198592
    //   sumsq   : 512 * 4             =     2048 B @ 25200640
    int*   offsets = (int*)(ws + 0);
    float* avg     = (float*)(ws + 32768);
    float* x       = (float*)(ws + 8421376);
    float* sums    = (float*)(ws + 25198592);
    float* sumsq   = (float*)(ws + 25200640);
    float* out     = (float*)d_out;

    // 1. exclusive scan of lengths
    excl_scan_kernel<<<1, 1024, 0, stream>>>(lengths, offsets);

    // 2. segment means (dominant 512MB stream, b128 coalesced)
    seg_mean_kernel<<<SEG_B, 64, 0, stream>>>(objs_attrs, lengths, offsets, avg);

    // 3. GEMM via fp32 WMMA: 128x16 = 2048 waves (64x32 region each), 4/block
    {
        const int totalWaves = (SEG_B / 64) * (DIM_E / 32);  // 2048
        const int wavesPerBlock = 4;
        gemm_wmma_kernel<<<totalWaves / wavesPerBlock, wavesPerBlock * 32, 0, stream>>>(
            avg, weight, bias, x);
    }

    // 4. BN stats
    zero_kernel<<<1, 1024, 0, stream>>>(sums, 2 * DIM_E);  // sums+sumsq contiguous
    {
        dim3 grid(SEG_B / 256, DIM_E / 256);  // 32 x 2
        bn_stats_kernel<<<grid, 256, 0, stream>>>(x, sums, sumsq);
    }

    // 5. BN apply -> d_out
    {
        const int total = SEG_B * DIM_E;  // 4M
        bn_apply_kernel<<<total / 256, 256, 0, stream>>>(x, sums, sumsq, gamma, beta, out);
    }
}